// GAT_40724879901272
// MI455X (gfx1250) — compile-verified
//
#include <hip/hip_runtime.h>
#include <math.h>

#define N_NODES 100000
#define N_EDGES 1600000
#define IN_DIM  128
#define HID     32
#define HEADS   4
#define OUT_DIM 16

typedef __attribute__((ext_vector_type(2))) float v2f;
typedef __attribute__((ext_vector_type(8))) float v8f;

// ---------- order-preserving float <-> uint map for atomic segment-max ----------
__device__ __forceinline__ unsigned fkey(float x) {
  unsigned b = __float_as_uint(x);
  return (b & 0x80000000u) ? ~b : (b | 0x80000000u);
}
__device__ __forceinline__ float funkey(unsigned k) {
  return __uint_as_float((k & 0x80000000u) ? (k & 0x7FFFFFFFu) : ~k);
}

// ---------- init ----------
__global__ void zero_f32(float* __restrict__ p, long long n) {
  long long t = (long long)blockIdx.x * blockDim.x + threadIdx.x;
  if (t < n) p[t] = 0.0f;
}

// ---------- strip-mined fp32 WMMA GEMM ----------
// C[M x NT*16] = A[M x K] * W[(NT*16) x K]^T.  One wave32 per 16-row strip:
// the A fragment is loaded once per k-step and reused across NT column tiles
// (NT accumulators, NT*8 VGPRs), cutting redundant A traffic NT-fold.
template <int NT>
__global__ void __launch_bounds__(256)
wmma_gemm_strip(const float* __restrict__ A, const float* __restrict__ W,
                float* __restrict__ C, int Mtiles, int K, int ldc, int coff) {
  const int wave = (blockIdx.x * blockDim.x + threadIdx.x) >> 5;
  if (wave >= Mtiles) return;
  const int lane = threadIdx.x & 31;
  const int lo = lane & 15;          // row (A) / col (B,D) within tile
  const int hi = lane >> 4;          // K-pair select (A,B) / M+8 select (D)
  const float* Ab = A + ((size_t)wave * 16 + lo) * K;
  v8f acc[NT];
#pragma unroll
  for (int j = 0; j < NT; ++j)
    acc[j] = (v8f){0.f, 0.f, 0.f, 0.f, 0.f, 0.f, 0.f, 0.f};
  for (int k0 = 0; k0 < K; k0 += 4) {
    const int ka = k0 + hi * 2;
    v2f a;
    a.x = Ab[ka]; a.y = Ab[ka + 1];
#pragma unroll
    for (int j = 0; j < NT; ++j) {
      const float* Wb = W + ((size_t)j * 16 + lo) * K;
      v2f b;
      b.x = Wb[ka]; b.y = Wb[ka + 1];
      acc[j] = __builtin_amdgcn_wmma_f32_16x16x4_f32(false, a, false, b,
                                                     (short)0, acc[j], false, false);
    }
  }
#pragma unroll
  for (int j = 0; j < NT; ++j) {
    float* Cb = C + (size_t)wave * 16 * ldc + coff + (size_t)j * 16 + lo;
#pragma unroll
    for (int r = 0; r < 8; ++r)
      Cb[(size_t)(r + hi * 8) * ldc] = acc[j][r];
  }
}

// ---------- gathered strip-mined WMMA GEMM for the frac-cover branch ----------
// hfpre[n, s*32 + o] = sum_i h[fc[n,s], i] * Wf[s][o][i]   (s = blockIdx.y)
__global__ void __launch_bounds__(256)
wmma_gemm_frac(const float* __restrict__ H, const int* __restrict__ fc,
               const float* __restrict__ Wf, float* __restrict__ hfpre) {
  const int wave = (blockIdx.x * blockDim.x + threadIdx.x) >> 5;
  if (wave >= N_NODES / 16) return;
  const int s = blockIdx.y;
  const int lane = threadIdx.x & 31;
  const int lo = lane & 15;
  const int hi = lane >> 4;
  const int nid = fc[(wave * 16 + lo) * 3 + s];        // gathered A row
  const float* Ab = H + (size_t)nid * IN_DIM;
  const float* Ws = Wf + (size_t)s * HID * IN_DIM;
  v8f acc[2];
  acc[0] = (v8f){0.f, 0.f, 0.f, 0.f, 0.f, 0.f, 0.f, 0.f};
  acc[1] = acc[0];
  for (int k0 = 0; k0 < IN_DIM; k0 += 4) {
    const int ka = k0 + hi * 2;
    v2f a;
    a.x = Ab[ka]; a.y = Ab[ka + 1];
#pragma unroll
    for (int j = 0; j < 2; ++j) {
      const float* Wb = Ws + ((size_t)j * 16 + lo) * IN_DIM;
      v2f b;
      b.x = Wb[ka]; b.y = Wb[ka + 1];
      acc[j] = __builtin_amdgcn_wmma_f32_16x16x4_f32(false, a, false, b,
                                                     (short)0, acc[j], false, false);
    }
  }
#pragma unroll
  for (int j = 0; j < 2; ++j) {
    float* Cb = hfpre + (size_t)wave * 16 * 96 + s * HID + (size_t)j * 16 + lo;
#pragma unroll
    for (int r = 0; r < 8; ++r)
      Cb[(size_t)(r + hi * 8) * 96] = acc[j][r];
  }
}

// ---------- per-node attention scores: s_src/s_dst = z . a_l / a_r ----------
__global__ void attn_scores(const float* __restrict__ z, const float* __restrict__ a,
                            float* __restrict__ ss, float* __restrict__ sd,
                            int n, int heads) {
  int t = blockIdx.x * blockDim.x + threadIdx.x;
  if (t >= n * heads) return;
  int node = t / heads, hh = t - node * heads;
  const float* zp = z + (size_t)node * heads * HID + hh * HID;
  const float* al = a + (size_t)hh * 2 * HID;
  const float* ar = al + HID;
  float s0 = 0.f, s1 = 0.f;
#pragma unroll 8
  for (int o = 0; o < HID; ++o) { s0 += zp[o] * al[o]; s1 += zp[o] * ar[o]; }
  ss[t] = s0; sd[t] = s1;
}

// ---------- edge pass 1: leaky-relu score + atomic segment max ----------
__global__ void edge_scores_max(const int* __restrict__ src, const int* __restrict__ dst,
                                const float* __restrict__ ss, const float* __restrict__ sd,
                                float* __restrict__ e, unsigned* __restrict__ m, int heads) {
  int t = blockIdx.x * blockDim.x + threadIdx.x;
  if (t >= N_EDGES * heads) return;
  int ed = t / heads, hh = t - ed * heads;
  float x = ss[src[ed] * heads + hh] + sd[dst[ed] * heads + hh];
  x = (x > 0.f) ? x : 0.01f * x;
  e[t] = x;
  atomicMax(m + dst[ed] * heads + hh, fkey(x));
}

// ---------- edge pass 2: exp(e - max) + atomic segment sum ----------
__global__ void edge_expsum(const int* __restrict__ dst, float* __restrict__ e,
                            const unsigned* __restrict__ m, float* __restrict__ sum,
                            int heads) {
  int t = blockIdx.x * blockDim.x + threadIdx.x;
  if (t >= N_EDGES * heads) return;
  int ed = t / heads, hh = t - ed * heads;
  int di = dst[ed] * heads + hh;
  float ee = expf(e[t] - funkey(m[di]));
  e[t] = ee;
  atomicAdd(sum + di, ee);
}

// ---------- edge pass 3: out[dst] += alpha * z[src] ----------
__global__ void edge_aggregate(const int* __restrict__ src, const int* __restrict__ dst,
                               const float* __restrict__ e, const float* __restrict__ sum,
                               const float* __restrict__ z, float* __restrict__ out,
                               int heads, int ldo) {
  long long t = (long long)blockIdx.x * blockDim.x + threadIdx.x;
  const int C = heads * HID;
  if (t >= (long long)N_EDGES * C) return;
  int ed = (int)(t / C);
  int c  = (int)(t - (long long)ed * C);
  int hh = c >> 5;                              // c / HID
  int di = dst[ed];
  float alpha = e[(size_t)ed * heads + hh] / fmaxf(sum[di * heads + hh], 1e-9f);
  atomicAdd(out + (size_t)di * ldo + c, alpha * z[(size_t)src[ed] * C + c]);
}

// ---------- elementwise ELU ----------
__global__ void elu_inplace(float* __restrict__ x, long long n) {
  long long t = (long long)blockIdx.x * blockDim.x + threadIdx.x;
  if (t >= n) return;
  float v = x[t];
  x[t] = (v > 0.f) ? v : (expf(v) - 1.0f);
}

extern "C" void kernel_launch(void* const* d_in, const int* in_sizes, int n_in,
                              void* d_out, int out_size, void* d_ws, size_t ws_size,
                              hipStream_t stream) {
  const float* h    = (const float*)d_in[0];
  const int*   src  = (const int*)  d_in[1];
  const int*   dst  = (const int*)  d_in[2];
  const int*   fc   = (const int*)  d_in[3];
  const float* W1   = (const float*)d_in[4];   // [4,32,128] -> flat [128,128]
  const float* a1   = (const float*)d_in[5];   // [4,64]
  const float* W2   = (const float*)d_in[6];   // [32,128]
  const float* a2   = (const float*)d_in[7];   // [64]
  const float* Wf   = (const float*)d_in[8];   // [3,32,128]
  const float* Wff  = (const float*)d_in[9];   // [32,96]
  const float* Wfc2 = (const float*)d_in[10];  // [16,64]
  float* out = (float*)d_out;

  float* w = (float*)d_ws;
  size_t off = 0;
  // zero-initialized accumulator region (contiguous)
  float*    agg1 = w + off;              off += (size_t)N_NODES * 128;   // h1 after ELU
  unsigned* m1   = (unsigned*)(w + off); off += (size_t)N_NODES * HEADS; // key(max), 0 == identity
  float*    sum1 = w + off;              off += (size_t)N_NODES * HEADS;
  unsigned* m2   = (unsigned*)(w + off); off += N_NODES;
  float*    sum2 = w + off;              off += N_NODES;
  float*    hg   = w + off;              off += (size_t)N_NODES * 64;    // [h2 | hf]
  const long long zeroN = (long long)off;
  // uninitialized scratch
  float* z1  = w + off; off += (size_t)N_NODES * 128;
  float* s1s = w + off; off += (size_t)N_NODES * HEADS;
  float* s1d = w + off; off += (size_t)N_NODES * HEADS;
  float* e1  = w + off; off += (size_t)N_EDGES * HEADS;
  float* z2  = w + off; off += (size_t)N_NODES * HID;
  float* s2s = w + off; off += N_NODES;
  float* s2d = w + off; off += N_NODES;
  float* e2  = w + off; off += N_EDGES;
  float* hfp = w + off; off += (size_t)N_NODES * 96;
  (void)ws_size; (void)in_sizes; (void)n_in; (void)out_size;

  const int TB = 256;
  const int MT = N_NODES / 16;                  // 6250 row tiles (exact)
  const int gemm_blocks = (MT + 7) / 8;         // 8 waves per block

  zero_f32<<<(int)((zeroN + TB - 1) / TB), TB, 0, stream>>>(w, zeroN);

  // ---- layer 1 : z1 = h @ W1^T  (K=128, 128 cols -> NT=8 strip) ----
  wmma_gemm_strip<8><<<gemm_blocks, TB, 0, stream>>>(h, W1, z1, MT, 128, 128, 0);
  attn_scores<<<(N_NODES * HEADS + TB - 1) / TB, TB, 0, stream>>>(z1, a1, s1s, s1d, N_NODES, HEADS);
  edge_scores_max<<<(N_EDGES * HEADS + TB - 1) / TB, TB, 0, stream>>>(src, dst, s1s, s1d, e1, m1, HEADS);
  edge_expsum<<<(N_EDGES * HEADS + TB - 1) / TB, TB, 0, stream>>>(dst, e1, m1, sum1, HEADS);
  {
    long long tot = (long long)N_EDGES * 128;
    edge_aggregate<<<(int)((tot + TB - 1) / TB), TB, 0, stream>>>(src, dst, e1, sum1, z1, agg1, HEADS, 128);
  }
  elu_inplace<<<(int)(((long long)N_NODES * 128 + TB - 1) / TB), TB, 0, stream>>>(
      agg1, (long long)N_NODES * 128);

  // ---- layer 2 (1 head) : z2 = h1 @ W2^T (NT=2); h2 accumulated into hg[:, 0:32] ----
  wmma_gemm_strip<2><<<gemm_blocks, TB, 0, stream>>>(agg1, W2, z2, MT, 128, HID, 0);
  attn_scores<<<(N_NODES + TB - 1) / TB, TB, 0, stream>>>(z2, a2, s2s, s2d, N_NODES, 1);
  edge_scores_max<<<(N_EDGES + TB - 1) / TB, TB, 0, stream>>>(src, dst, s2s, s2d, e2, m2, 1);
  edge_expsum<<<(N_EDGES + TB - 1) / TB, TB, 0, stream>>>(dst, e2, m2, sum2, 1);
  {
    long long tot = (long long)N_EDGES * HID;
    edge_aggregate<<<(int)((tot + TB - 1) / TB), TB, 0, stream>>>(src, dst, e2, sum2, z2, hg, 1, 64);
  }

  // ---- frac-cover branch: hfpre[N,96] (gathered GEMM) then hf -> hg[:, 32:64] ----
  {
    dim3 g(gemm_blocks, 3);
    wmma_gemm_frac<<<g, TB, 0, stream>>>(h, fc, Wf, hfp);
  }
  wmma_gemm_strip<2><<<gemm_blocks, TB, 0, stream>>>(hfp, Wff, hg, MT, 96, 64, HID);

  // ---- final projection: out[N,16] = hg[N,64] @ Wfc2^T (NT=1) ----
  wmma_gemm_strip<1><<<gemm_blocks, TB, 0, stream>>>(hg, Wfc2, out, MT, 64, OUT_DIM, 0);
}